// AdditiveAttentionLayer_31482110280201
// MI455X (gfx1250) — compile-verified
//
#include <hip/hip_runtime.h>
#include <hip/hip_bf16.h>
#include <math.h>

// AdditiveAttention fused pipeline for MI455X (gfx1250, wave32).
//   GEMMs: V_WMMA_F32_16X16X4_F32, 128x64 block tile, B panel double-buffered
//          in LDS via the Tensor Data Mover (tensor_load_to_lds, TENSORcnt).
//   Scores: native v_tanh_f32 (TRANS pipe); (B,Q,K,H) never materialized.

typedef __attribute__((ext_vector_type(2))) float v2f;
typedef __attribute__((ext_vector_type(8))) float v8f;

#define NEGC (-1000000.0f)

#if defined(__has_builtin)
#  if __has_builtin(__builtin_amdgcn_tanhf)
#    define FAST_TANH(x) __builtin_amdgcn_tanhf(x)
#  else
#    define FAST_TANH(x) tanhf(x)
#  endif
#else
#  define FAST_TANH(x) tanhf(x)
#endif

#ifndef USE_TDM
#define USE_TDM 1
#endif
#if USE_TDM && defined(__has_builtin)
#  if __has_builtin(__builtin_amdgcn_tensor_load_to_lds)
#    define HAVE_TDM 1
#  endif
#endif
#ifndef HAVE_TDM
#  define HAVE_TDM 0
#endif

#if HAVE_TDM
typedef unsigned int v4u __attribute__((ext_vector_type(4)));
typedef int          v8i __attribute__((ext_vector_type(8)));
typedef int          v4i __attribute__((ext_vector_type(4)));

// Issue one TDM 2D tile load: rows x cols fp32 tile at gsrc (row stride =
// row_stride elements) -> LDS at lds_byte_off (packed row-major, cols wide).
// D# layout per CDNA5 ISA. 6-arg builtin form:
//   (uint32x4 g0, int32x8 g1, int32x4 g2, int32x4 g3, int32x8 ext, i32 cpol)
// groups 2/3/ext unused for a 2D tile.
__device__ __forceinline__ void tdm_load_tile_2d(const float* gsrc,
                                                 unsigned lds_byte_off,
                                                 int rows, int cols,
                                                 int row_stride)
{
    unsigned long long ga = (unsigned long long)(uintptr_t)gsrc;
    v4u g0;
    g0[0] = 1u;                                    // count=1 (valid descriptor)
    g0[1] = lds_byte_off;                          // lds_addr
    g0[2] = (unsigned)(ga & 0xFFFFFFFFu);          // global_addr[31:0]
    g0[3] = (unsigned)((ga >> 32) & 0x1FFFFFFu)    // global_addr[56:32]
          | (2u << 30);                            // type=2 ("image")
    unsigned td0 = (unsigned)row_stride;           // tensor_dim0 (>= tile: no OOB)
    unsigned td1 = (unsigned)rows;                 // tensor_dim1
    unsigned long long st0 = (unsigned long long)row_stride;
    v8i g1;
    g1[0] = (int)(2u << 16);                       // data_size=2 -> 4 bytes
    g1[1] = (int)((td0 & 0xFFFFu) << 16);          // tensor_dim0[15:0] @ bits 63:48
    g1[2] = (int)((td0 >> 16) | ((td1 & 0xFFFFu) << 16));
    g1[3] = (int)((td1 >> 16) | ((unsigned)cols << 16));   // tile_dim0
    g1[4] = (int)((unsigned)rows & 0xFFFFu);       // tile_dim1 (tile_dim2=0)
    g1[5] = (int)(st0 & 0xFFFFFFFFu);              // tensor_dim0_stride[31:0]
    g1[6] = (int)((st0 >> 32) & 0xFFFFu);          // stride0[47:32], stride1 lo=0
    g1[7] = 0;
    v4i z4 = {};
    v8i z8 = {};
    __builtin_amdgcn_tensor_load_to_lds(g0, g1, z4, z4, z8, 0);
}
#endif

// ---------------------------------------------------------------------------
// fp32 WMMA GEMM, block tile 128x64:
//   grid.x = N/64, grid.y = M/128, grid.z = batch
//   8 waves; wave w owns rows [w*16, w*16+16), 4 N-subtiles of 16.
//   K chunked by 64; B chunk (64x64 f32, 16 KB) double-buffered in LDS: wave 0
//   issues the TDM load of chunk c+1 before s_wait_tensorcnt on chunk c (TDM
//   ops of one wave complete in order), so the DMA overlaps the 16 WMMA steps.
//   A read as float2 per k-step, reused across 4 WMMAs -> ~21 FLOP/B global.
// WMMA f32 16x16x4 lane layout (ISA 7.12.2):
//   A: lane&15 = M row, K = (lane>>4)*2 + vgpr
//   B: lane&15 = N col, K = (lane>>4)*2 + vgpr
//   C: row = (lane>>4)*8 + vgpr, col = lane&15
// ---------------------------------------------------------------------------
#define KCHUNK 64

__global__ __launch_bounds__(256) void wmma_gemm_f32(
    const float* __restrict__ A, const float* __restrict__ B,
    float* __restrict__ C,
    int K, int lda, int ldb, int ldc,
    long long asb, long long bsb, long long csb)
{
    const int batch = blockIdx.z;
    A += batch * asb;
    B += batch * bsb;
    C += batch * csb;

    __shared__ float Bsh[2][KCHUNK * 64];          // 2 x 16 KB

    const int tid  = threadIdx.x;
    const int wave = tid >> 5;
    const int lane = tid & 31;
    const int half = lane >> 4;                    // selects K pair (0/1)
    const int l15  = lane & 15;

    const int m0 = blockIdx.y * 128 + wave * 16;   // wave's row block
    const int nb = blockIdx.x * 64;                // block's col block

    const float* Ar = A + (long long)(m0 + l15) * lda + half * 2;
    const int nchunks = K / KCHUNK;

    v8f acc0 = {}, acc1 = {}, acc2 = {}, acc3 = {};

#if HAVE_TDM
    if (wave == 0)                                 // prologue: chunk 0 -> buf 0
        tdm_load_tile_2d(B + nb, (unsigned)(uintptr_t)&Bsh[0][0],
                         KCHUNK, 64, ldb);
#endif

    for (int c = 0; c < nchunks; ++c) {
        const int k0 = c * KCHUNK;
        const float* Bs = &Bsh[c & 1][0];
#if HAVE_TDM
        if (wave == 0) {
            if (c + 1 < nchunks) {                 // prefetch next chunk
                tdm_load_tile_2d(B + (long long)(k0 + KCHUNK) * ldb + nb,
                                 (unsigned)(uintptr_t)&Bsh[(c + 1) & 1][0],
                                 KCHUNK, 64, ldb);
                __builtin_amdgcn_s_wait_tensorcnt(1);  // chunk c complete
            } else {
                __builtin_amdgcn_s_wait_tensorcnt(0);
            }
        }
        __syncthreads();
#else
        for (int i = tid; i < KCHUNK * 16; i += 256) {       // 1024 float4s
            int row = i >> 4, c4 = (i & 15) << 2;
            *(float4*)&Bsh[c & 1][row * 64 + c4] =
                *(const float4*)&B[(long long)(k0 + row) * ldb + nb + c4];
        }
        __syncthreads();
#endif
        // ---- 16 WMMA k-steps over the chunk ----
#pragma unroll 4
        for (int kk = 0; kk < KCHUNK; kk += 4) {
            const int kr = kk + half * 2;
            float2 a2 = *(const float2*)(Ar + k0 + kk);
            v2f av; av.x = a2.x; av.y = a2.y;
            v2f b0, b1, b2, b3;
            b0.x = Bs[kr * 64 + l15];       b0.y = Bs[(kr + 1) * 64 + l15];
            b1.x = Bs[kr * 64 + 16 + l15];  b1.y = Bs[(kr + 1) * 64 + 16 + l15];
            b2.x = Bs[kr * 64 + 32 + l15];  b2.y = Bs[(kr + 1) * 64 + 32 + l15];
            b3.x = Bs[kr * 64 + 48 + l15];  b3.y = Bs[(kr + 1) * 64 + 48 + l15];
            acc0 = __builtin_amdgcn_wmma_f32_16x16x4_f32(false, av, false, b0, (short)0, acc0, false, false);
            acc1 = __builtin_amdgcn_wmma_f32_16x16x4_f32(false, av, false, b1, (short)0, acc1, false, false);
            acc2 = __builtin_amdgcn_wmma_f32_16x16x4_f32(false, av, false, b2, (short)0, acc2, false, false);
            acc3 = __builtin_amdgcn_wmma_f32_16x16x4_f32(false, av, false, b3, (short)0, acc3, false, false);
        }
        __syncthreads();       // readers done before buf[c&1] is re-filled
    }

    float* Cp = C + (long long)(m0 + half * 8) * ldc + nb + l15;
#pragma unroll
    for (int r = 0; r < 8; ++r) {
        Cp[(long long)r * ldc +  0] = acc0[r];
        Cp[(long long)r * ldc + 16] = acc1[r];
        Cp[(long long)r * ldc + 32] = acc2[r];
        Cp[(long long)r * ldc + 48] = acc3[r];
    }
}

// ---------------------------------------------------------------------------
// Fused score + mask + softmax for one (b,q) row:
// TRANS-pipe bound; masked k skipped (halves tanh work on average).
// ---------------------------------------------------------------------------
#define HDIM 256
#define KDIM 1024

__global__ __launch_bounds__(256) void score_softmax(
    const float* __restrict__ qh, const float* __restrict__ kh,
    const float* __restrict__ wv, const int* __restrict__ valid,
    float* __restrict__ attn, int Qn)
{
    const int bq   = blockIdx.x;
    const int b    = bq / Qn;
    const int tid  = threadIdx.x;
    const int wave = tid >> 5;
    const int lane = tid & 31;

    __shared__ float qrow[HDIM];
    __shared__ float wls[HDIM];
    __shared__ float sc[KDIM];
    __shared__ float red[8];

    qrow[tid] = qh[(long long)bq * HDIM + tid];
    wls[tid]  = wv[tid];
    const int vlen = valid[b];
    __syncthreads();

    for (int k = wave; k < KDIM; k += 8) {
        float s;
        if (k < vlen) {                          // wave-uniform branch
            const float4* k4 = reinterpret_cast<const float4*>(
                kh + ((long long)b * KDIM + k) * HDIM);
            float p = 0.f;
#pragma unroll
            for (int h4 = lane; h4 < HDIM / 4; h4 += 32) {
                float4 kv = k4[h4];
                int h = h4 * 4;
                p += FAST_TANH(qrow[h + 0] + kv.x) * wls[h + 0];
                p += FAST_TANH(qrow[h + 1] + kv.y) * wls[h + 1];
                p += FAST_TANH(qrow[h + 2] + kv.z) * wls[h + 2];
                p += FAST_TANH(qrow[h + 3] + kv.w) * wls[h + 3];
            }
            for (int off = 16; off; off >>= 1) p += __shfl_xor(p, off, 32);
            s = p;
        } else {
            s = NEGC;
        }
        if (lane == 0) sc[k] = s;
    }
    __syncthreads();

    float m = -INFINITY;
    for (int k = tid; k < KDIM; k += 256) m = fmaxf(m, sc[k]);
    for (int off = 16; off; off >>= 1) m = fmaxf(m, __shfl_xor(m, off, 32));
    if (lane == 0) red[wave] = m;
    __syncthreads();
    m = red[0];
#pragma unroll
    for (int i = 1; i < 8; ++i) m = fmaxf(m, red[i]);
    __syncthreads();

    float sum = 0.f;
    for (int k = tid; k < KDIM; k += 256) {
        float e = __expf(sc[k] - m);             // masked: underflows to 0
        sc[k] = e;
        sum += e;
    }
    for (int off = 16; off; off >>= 1) sum += __shfl_xor(sum, off, 32);
    if (lane == 0) red[wave] = sum;
    __syncthreads();
    sum = red[0];
#pragma unroll
    for (int i = 1; i < 8; ++i) sum += red[i];
    const float inv = 1.f / sum;

    for (int k = tid; k < KDIM; k += 256)        // re-reads own writes only
        attn[(long long)bq * KDIM + k] = sc[k] * inv;
}

// ---------------------------------------------------------------------------
// Launch: inputs = {queries, keys, values, valid_lens, Wq, Wk, wv}
// ---------------------------------------------------------------------------
extern "C" void kernel_launch(void* const* d_in, const int* in_sizes, int n_in,
                              void* d_out, int out_size, void* d_ws, size_t ws_size,
                              hipStream_t stream)
{
    const float* queries = (const float*)d_in[0];  // (8,128,512)
    const float* keys    = (const float*)d_in[1];  // (8,1024,512)
    const float* values  = (const float*)d_in[2];  // (8,1024,512)
    const int*   vlens   = (const int*)  d_in[3];  // (8,)
    const float* Wq      = (const float*)d_in[4];  // (512,256)
    const float* Wk      = (const float*)d_in[5];  // (512,256)
    const float* wv      = (const float*)d_in[6];  // (256,)
    float* out = (float*)d_out;                    // (8,128,512)

    const int Bb = 8, Qn = 128, Kk = 1024, D = 512, Hh = 256;

    float* qh   = (float*)d_ws;                    // 1024*256   (1 MB)
    float* khp  = qh  + (size_t)Bb * Qn * Hh;      // 8192*256   (8 MB)
    float* attn = khp + (size_t)Bb * Kk * Hh;      // 1024*1024  (4 MB)

    // 1) q projection: (1024 x 512) * (512 x 256)
    wmma_gemm_f32<<<dim3(Hh / 64, (Bb * Qn) / 128, 1), 256, 0, stream>>>(
        queries, Wq, qh, D, D, Hh, Hh, 0, 0, 0);

    // 2) k projection: (8192 x 512) * (512 x 256)
    wmma_gemm_f32<<<dim3(Hh / 64, (Bb * Kk) / 128, 1), 256, 0, stream>>>(
        keys, Wk, khp, D, D, Hh, Hh, 0, 0, 0);

    // 3) fused tanh-score + mask + softmax
    score_softmax<<<Bb * Qn, 256, 0, stream>>>(qh, khp, wv, vlens, attn, Qn);

    // 4) out[b] = attn[b] (128 x 1024) * values[b] (1024 x 512), batched
    wmma_gemm_f32<<<dim3(D / 64, Qn / 128, Bb), 256, 0, stream>>>(
        attn, values, out, Kk, Kk, D, D,
        (long long)Qn * Kk, (long long)Kk * D, (long long)Qn * D);
}